// STGCNTurbulence_40321152974903
// MI455X (gfx1250) — compile-verified
//
#include <hip/hip_runtime.h>
#include <math.h>

#define N_NODES 1024
#define T_STEPS 10
#define C_CH    64
#define KW      3
#define F_S     9
#define F_I     19
#define F_ALL   28

typedef __attribute__((ext_vector_type(2))) float v2f;
typedef __attribute__((ext_vector_type(8))) float v8f;

__device__ __forceinline__ float gelu_exact(float x) {
  return 0.5f * x * (1.0f + erff(x * 0.70710678118654752440f));
}

__device__ __forceinline__ float wave_sum(float v) {
#pragma unroll
  for (int m = 16; m > 0; m >>= 1) v += __shfl_xor(v, m, 32);
  return v;
}

// ---------------------------------------------------------------------------
// Kernel 1: soft adjacency W[i,j] + symmetric-norm scale dm[i]
// W is symmetric and W[i,i] = 1, so the full row sum equals deg[i]
// (pairwise sum over s!=i plus the self-loop weight 1).
// ---------------------------------------------------------------------------
__global__ void __launch_bounds__(256) k_adj_dm(const float* __restrict__ pos,
                                                float* __restrict__ Wm,
                                                float* __restrict__ dm) {
  const int i = blockIdx.x;
  const int tid = threadIdx.x;
  const float pix = pos[i * 3 + 0], piy = pos[i * 3 + 1], piz = pos[i * 3 + 2];
  const float invH = 1.0f / (2.0f * 5000.0f * 5000.0f);
  const float invV = 1.0f / (2.0f * 300.0f * 300.0f);
  float part = 0.0f;
  for (int j = tid; j < N_NODES; j += 256) {
    const float dx = pix - pos[j * 3 + 0];
    const float dy = piy - pos[j * 3 + 1];
    const float dz = piz - pos[j * 3 + 2];
    const float w = __expf(-(dx * dx + dy * dy) * invH - dz * dz * invV);
    Wm[(size_t)i * N_NODES + j] = w;
    part += w;
  }
  __shared__ float red[8];
  const float wsum = wave_sum(part);
  const int lane = tid & 31, wv = tid >> 5;
  if (lane == 0) red[wv] = wsum;
  __syncthreads();
  if (tid == 0) {
    float deg = 0.0f;
#pragma unroll
    for (int k = 0; k < 8; ++k) deg += red[k];
    dm[i] = fminf(rsqrtf(deg), 1e4f);
  }
}

// ---------------------------------------------------------------------------
// Kernel 2: input projection: gelu(LN(x_s @ sp_w^T)) + gelu(LN(x_i @ ip_w^T))
// One wave32 per (n,t) row, 2 channels per lane, LN via wave butterfly.
// ---------------------------------------------------------------------------
__global__ void __launch_bounds__(256) k_inproj(
    const float* __restrict__ x,
    const float* __restrict__ sp_w, const float* __restrict__ sp_b,
    const float* __restrict__ sp_g, const float* __restrict__ sp_be,
    const float* __restrict__ ip_w, const float* __restrict__ ip_b,
    const float* __restrict__ ip_g, const float* __restrict__ ip_be,
    float* __restrict__ e) {
  const int lane = threadIdx.x & 31;
  const int wv = threadIdx.x >> 5;
  const int row = blockIdx.x * 8 + wv;  // [0, N*T)
  const float* xr = x + (size_t)row * F_ALL;
  float xin[F_ALL];
#pragma unroll
  for (int k = 0; k < F_ALL; ++k) xin[k] = xr[k];
  const int c0 = lane, c1 = lane + 32;

  float s0 = sp_b[c0], s1 = sp_b[c1];
#pragma unroll
  for (int k = 0; k < F_S; ++k) {
    s0 = fmaf(xin[k], sp_w[c0 * F_S + k], s0);
    s1 = fmaf(xin[k], sp_w[c1 * F_S + k], s1);
  }
  float i0 = ip_b[c0], i1 = ip_b[c1];
#pragma unroll
  for (int k = 0; k < F_I; ++k) {
    i0 = fmaf(xin[F_S + k], ip_w[c0 * F_I + k], i0);
    i1 = fmaf(xin[F_S + k], ip_w[c1 * F_I + k], i1);
  }
  // LN + GELU, sensor branch
  float m = wave_sum(s0 + s1) * (1.0f / 64.0f);
  float d0 = s0 - m, d1 = s1 - m;
  float var = wave_sum(d0 * d0 + d1 * d1) * (1.0f / 64.0f);
  float inv = rsqrtf(var + 1e-5f);
  const float a0 = gelu_exact(d0 * inv * sp_g[c0] + sp_be[c0]);
  const float a1 = gelu_exact(d1 * inv * sp_g[c1] + sp_be[c1]);
  // LN + GELU, index branch
  m = wave_sum(i0 + i1) * (1.0f / 64.0f);
  d0 = i0 - m; d1 = i1 - m;
  var = wave_sum(d0 * d0 + d1 * d1) * (1.0f / 64.0f);
  inv = rsqrtf(var + 1e-5f);
  const float b0 = gelu_exact(d0 * inv * ip_g[c0] + ip_be[c0]);
  const float b1 = gelu_exact(d1 * inv * ip_g[c1] + ip_be[c1]);

  e[(size_t)row * C_CH + c0] = a0 + b0;
  e[(size_t)row * C_CH + c1] = a1 + b1;
}

// ---------------------------------------------------------------------------
// Kernel 3: two causal conv blocks + residual + temporal mean pool, fused.
// One block per node; e/h tiles staged in LDS (~7.5 KB of the 320 KB WGP LDS).
// Emits x_pool and xs = dm[n] * x_pool (pre-scaled GEMM operand).
// ---------------------------------------------------------------------------
__global__ void __launch_bounds__(256) k_conv_pool(
    const float* __restrict__ e,
    const float* __restrict__ c1w, const float* __restrict__ c1b,
    const float* __restrict__ b1g, const float* __restrict__ b1b,
    const float* __restrict__ b1m, const float* __restrict__ b1v,
    const float* __restrict__ c2w, const float* __restrict__ c2b,
    const float* __restrict__ b2g, const float* __restrict__ b2b,
    const float* __restrict__ b2m, const float* __restrict__ b2v,
    const float* __restrict__ dm,
    float* __restrict__ x_pool, float* __restrict__ xs) {
  __shared__ float eL[T_STEPS][C_CH];
  __shared__ float hA[T_STEPS][C_CH];
  __shared__ float hB[T_STEPS][C_CH];
  const int n = blockIdx.x;
  const int tid = threadIdx.x;

  for (int idx = tid; idx < T_STEPS * C_CH; idx += 256)
    eL[idx / C_CH][idx % C_CH] = e[(size_t)n * T_STEPS * C_CH + idx];
  __syncthreads();

  // conv block 1 (causal left pad K-1)
  for (int idx = tid; idx < T_STEPS * C_CH; idx += 256) {
    const int t = idx / C_CH, co = idx % C_CH;
    float acc = c1b[co];
#pragma unroll
    for (int k = 0; k < KW; ++k) {
      const int ti = t + k - (KW - 1);
      if (ti >= 0) {
        const float* wrow = c1w + (size_t)co * C_CH * KW + k;
#pragma unroll 8
        for (int ci = 0; ci < C_CH; ++ci)
          acc = fmaf(eL[ti][ci], wrow[ci * KW], acc);
      }
    }
    acc = (acc - b1m[co]) * rsqrtf(b1v[co] + 1e-5f) * b1g[co] + b1b[co];
    hA[t][co] = gelu_exact(acc);
  }
  __syncthreads();

  // conv block 2
  for (int idx = tid; idx < T_STEPS * C_CH; idx += 256) {
    const int t = idx / C_CH, co = idx % C_CH;
    float acc = c2b[co];
#pragma unroll
    for (int k = 0; k < KW; ++k) {
      const int ti = t + k - (KW - 1);
      if (ti >= 0) {
        const float* wrow = c2w + (size_t)co * C_CH * KW + k;
#pragma unroll 8
        for (int ci = 0; ci < C_CH; ++ci)
          acc = fmaf(hA[ti][ci], wrow[ci * KW], acc);
      }
    }
    acc = (acc - b2m[co]) * rsqrtf(b2v[co] + 1e-5f) * b2g[co] + b2b[co];
    hB[t][co] = gelu_exact(acc);
  }
  __syncthreads();

  // residual + temporal mean pool
  if (tid < C_CH) {
    float s = 0.0f;
#pragma unroll
    for (int t = 0; t < T_STEPS; ++t) s += hB[t][tid] + eL[t][tid];
    s *= (1.0f / (float)T_STEPS);
    x_pool[(size_t)n * C_CH + tid] = s;
    xs[(size_t)n * C_CH + tid] = s * dm[n];
  }
}

// ---------------------------------------------------------------------------
// Kernel 4: dense GCN aggregation as fp32 WMMA GEMM:
//   D[1024,64] = Wm[1024,1024] @ B[1024,64]   (B already scaled by dm[s])
// One wave per 16x16 output tile; V_WMMA_F32_16X16X4_F32 over K in steps of 4.
// A frag (16x4 f32): lane M = lane&15; VGPR0/1 hold K = 2*(lane>>4), +1.
// D layout: acc[j] = D[m0 + j + 8*(lane>>4)][n0 + (lane&15)].
// ---------------------------------------------------------------------------
__global__ void __launch_bounds__(256) k_gcn_agg_wmma(
    const float* __restrict__ A, const float* __restrict__ B,
    float* __restrict__ D) {
  const int lane = threadIdx.x & 31;
  const int wv = threadIdx.x >> 5;
  const int gid = blockIdx.x * 8 + wv;  // 0..255 : 64 M-tiles x 4 N-tiles
  const int m0 = (gid >> 2) * 16;
  const int n0 = (gid & 3) * 16;
  const int r = lane & 15;
  const int kh = lane >> 4;  // 0 or 1

  const float* Arow = A + (size_t)(m0 + r) * N_NODES + 2 * kh;
  const float* Bcol = B + n0 + r + (size_t)(2 * kh) * C_CH;

  v8f acc = {};
#pragma unroll 4
  for (int k = 0; k < N_NODES; k += 4) {
    v2f a, b;
    const float2 av = *(const float2*)(Arow + k);
    a.x = av.x;
    a.y = av.y;
    b.x = Bcol[(size_t)k * C_CH];
    b.y = Bcol[(size_t)k * C_CH + C_CH];
    acc = __builtin_amdgcn_wmma_f32_16x16x4_f32(
        /*neg_a=*/false, a, /*neg_b=*/false, b,
        /*c_mod=*/(short)0, acc, /*reuse_a=*/false, /*reuse_b=*/false);
  }
#pragma unroll
  for (int j = 0; j < 8; ++j)
    D[(size_t)(m0 + j + 8 * kh) * C_CH + n0 + r] = acc[j];
}

// ---------------------------------------------------------------------------
// Kernel 5: GCN post: h_out = gelu(LN(dm[d]*agg @ W^T + b + h_prev))
// One wave32 per row; agg row staged in LDS; optionally emits dm-scaled copy.
// ---------------------------------------------------------------------------
__global__ void __launch_bounds__(256) k_gcn_post(
    const float* __restrict__ agg, const float* __restrict__ hPrev,
    const float* __restrict__ gw, const float* __restrict__ gb,
    const float* __restrict__ gg, const float* __restrict__ gbe,
    const float* __restrict__ dm,
    float* __restrict__ hOut, float* __restrict__ hsOut) {
  __shared__ float rowb[8][C_CH];
  const int lane = threadIdx.x & 31;
  const int wv = threadIdx.x >> 5;
  const int r = blockIdx.x * 8 + wv;
  const float d = dm[r];
  rowb[wv][lane] = agg[(size_t)r * C_CH + lane] * d;
  rowb[wv][lane + 32] = agg[(size_t)r * C_CH + lane + 32] * d;
  __syncthreads();

  const int c0 = lane, c1 = lane + 32;
  float a0 = gb[c0] + hPrev[(size_t)r * C_CH + c0];
  float a1 = gb[c1] + hPrev[(size_t)r * C_CH + c1];
#pragma unroll 8
  for (int ci = 0; ci < C_CH; ++ci) {
    const float hv = rowb[wv][ci];
    a0 = fmaf(hv, gw[c0 * C_CH + ci], a0);
    a1 = fmaf(hv, gw[c1 * C_CH + ci], a1);
  }
  const float m = wave_sum(a0 + a1) * (1.0f / 64.0f);
  const float d0 = a0 - m, d1 = a1 - m;
  const float var = wave_sum(d0 * d0 + d1 * d1) * (1.0f / 64.0f);
  const float inv = rsqrtf(var + 1e-5f);
  const float y0 = gelu_exact(d0 * inv * gg[c0] + gbe[c0]);
  const float y1 = gelu_exact(d1 * inv * gg[c1] + gbe[c1]);
  hOut[(size_t)r * C_CH + c0] = y0;
  hOut[(size_t)r * C_CH + c1] = y1;
  if (hsOut) {
    hsOut[(size_t)r * C_CH + c0] = y0 * d;
    hsOut[(size_t)r * C_CH + c1] = y1 * d;
  }
}

// ---------------------------------------------------------------------------
// Kernel 6: dual-head MLP, fully fused per row.
// out[0 : 27N) = mu flat, out[27N : 54N) = log_var flat.
// ---------------------------------------------------------------------------
__global__ void __launch_bounds__(256) k_head(
    const float* __restrict__ x_pool, const float* __restrict__ h2,
    const float* __restrict__ t1w, const float* __restrict__ t1b,
    const float* __restrict__ t1g, const float* __restrict__ t1be,
    const float* __restrict__ t2w, const float* __restrict__ t2b,
    const float* __restrict__ t2g, const float* __restrict__ t2be,
    const float* __restrict__ muw, const float* __restrict__ mub,
    const float* __restrict__ lvw, const float* __restrict__ lvb,
    float* __restrict__ out) {
  __shared__ float buf[8][2 * C_CH];
  const int lane = threadIdx.x & 31;
  const int wv = threadIdx.x >> 5;
  const int r = blockIdx.x * 8 + wv;
  const int c0 = lane, c1 = lane + 32;

  buf[wv][lane] = x_pool[(size_t)r * C_CH + lane];
  buf[wv][lane + 32] = x_pool[(size_t)r * C_CH + lane + 32];
  buf[wv][64 + lane] = h2[(size_t)r * C_CH + lane];
  buf[wv][96 + lane] = h2[(size_t)r * C_CH + lane + 32];
  __syncthreads();

  // layer t1: 128 -> 64
  float a0 = t1b[c0], a1 = t1b[c1];
#pragma unroll 8
  for (int ci = 0; ci < 2 * C_CH; ++ci) {
    const float fv = buf[wv][ci];
    a0 = fmaf(fv, t1w[c0 * 2 * C_CH + ci], a0);
    a1 = fmaf(fv, t1w[c1 * 2 * C_CH + ci], a1);
  }
  {
    const float m = wave_sum(a0 + a1) * (1.0f / 64.0f);
    const float d0 = a0 - m, d1 = a1 - m;
    const float var = wave_sum(d0 * d0 + d1 * d1) * (1.0f / 64.0f);
    const float inv = rsqrtf(var + 1e-5f);
    a0 = gelu_exact(d0 * inv * t1g[c0] + t1be[c0]);
    a1 = gelu_exact(d1 * inv * t1g[c1] + t1be[c1]);
  }
  __syncthreads();
  buf[wv][c0] = a0;
  buf[wv][c1] = a1;
  __syncthreads();

  // layer t2: 64 -> 64
  a0 = t2b[c0];
  a1 = t2b[c1];
#pragma unroll 8
  for (int ci = 0; ci < C_CH; ++ci) {
    const float fv = buf[wv][ci];
    a0 = fmaf(fv, t2w[c0 * C_CH + ci], a0);
    a1 = fmaf(fv, t2w[c1 * C_CH + ci], a1);
  }
  {
    const float m = wave_sum(a0 + a1) * (1.0f / 64.0f);
    const float d0 = a0 - m, d1 = a1 - m;
    const float var = wave_sum(d0 * d0 + d1 * d1) * (1.0f / 64.0f);
    const float inv = rsqrtf(var + 1e-5f);
    a0 = gelu_exact(d0 * inv * t2g[c0] + t2be[c0]);
    a1 = gelu_exact(d1 * inv * t2g[c1] + t2be[c1]);
  }
  __syncthreads();
  buf[wv][c0] = a0;
  buf[wv][c1] = a1;
  __syncthreads();

  // heads: 27 outputs each
  for (int o = lane; o < 27; o += 32) {
    float mu = mub[o], lv = lvb[o];
#pragma unroll 8
    for (int ci = 0; ci < C_CH; ++ci) {
      const float sv = buf[wv][ci];
      mu = fmaf(sv, muw[o * C_CH + ci], mu);
      lv = fmaf(sv, lvw[o * C_CH + ci], lv);
    }
    const float z = 5.0f * mu;
    const float sp = fmaxf(z, 0.0f) + log1pf(expf(-fabsf(z)));  // stable softplus
    out[(size_t)r * 27 + o] = sp * 0.2f;
    out[(size_t)27 * N_NODES + (size_t)r * 27 + o] =
        fminf(fmaxf(lv, -6.0f), 4.0f);
  }
}

// ---------------------------------------------------------------------------
extern "C" void kernel_launch(void* const* d_in, const int* in_sizes, int n_in,
                              void* d_out, int out_size, void* d_ws,
                              size_t ws_size, hipStream_t stream) {
  (void)in_sizes; (void)n_in; (void)out_size; (void)ws_size;
  const float* x     = (const float*)d_in[0];
  const float* pos   = (const float*)d_in[1];
  const float* sp_w  = (const float*)d_in[2];
  const float* sp_b  = (const float*)d_in[3];
  const float* sp_g  = (const float*)d_in[4];
  const float* sp_be = (const float*)d_in[5];
  const float* ip_w  = (const float*)d_in[6];
  const float* ip_b  = (const float*)d_in[7];
  const float* ip_g  = (const float*)d_in[8];
  const float* ip_be = (const float*)d_in[9];
  const float* c1_w  = (const float*)d_in[10];
  const float* c1_b  = (const float*)d_in[11];
  const float* bn1_g = (const float*)d_in[12];
  const float* bn1_b = (const float*)d_in[13];
  const float* bn1_m = (const float*)d_in[14];
  const float* bn1_v = (const float*)d_in[15];
  const float* c2_w  = (const float*)d_in[16];
  const float* c2_b  = (const float*)d_in[17];
  const float* bn2_g = (const float*)d_in[18];
  const float* bn2_b = (const float*)d_in[19];
  const float* bn2_m = (const float*)d_in[20];
  const float* bn2_v = (const float*)d_in[21];
  const float* g1_w  = (const float*)d_in[22];
  const float* g1_b  = (const float*)d_in[23];
  const float* g1_g  = (const float*)d_in[24];
  const float* g1_be = (const float*)d_in[25];
  const float* g2_w  = (const float*)d_in[26];
  const float* g2_b  = (const float*)d_in[27];
  const float* g2_g  = (const float*)d_in[28];
  const float* g2_be = (const float*)d_in[29];
  const float* t1_w  = (const float*)d_in[30];
  const float* t1_b  = (const float*)d_in[31];
  const float* t1_g  = (const float*)d_in[32];
  const float* t1_be = (const float*)d_in[33];
  const float* t2_w  = (const float*)d_in[34];
  const float* t2_b  = (const float*)d_in[35];
  const float* t2_g  = (const float*)d_in[36];
  const float* t2_be = (const float*)d_in[37];
  const float* mu_w  = (const float*)d_in[38];
  const float* mu_b  = (const float*)d_in[39];
  const float* lv_w  = (const float*)d_in[40];
  const float* lv_b  = (const float*)d_in[41];

  float* ws = (float*)d_ws;
  size_t off = 0;
  float* Wm     = ws + off; off += (size_t)N_NODES * N_NODES;     // 4 MB
  float* dm     = ws + off; off += N_NODES;
  float* e      = ws + off; off += (size_t)N_NODES * T_STEPS * C_CH;
  float* x_pool = ws + off; off += (size_t)N_NODES * C_CH;
  float* xs     = ws + off; off += (size_t)N_NODES * C_CH;
  float* agg    = ws + off; off += (size_t)N_NODES * C_CH;
  float* h1     = ws + off; off += (size_t)N_NODES * C_CH;
  float* hs1    = ws + off; off += (size_t)N_NODES * C_CH;
  float* h2     = ws + off; off += (size_t)N_NODES * C_CH;

  k_adj_dm<<<N_NODES, 256, 0, stream>>>(pos, Wm, dm);
  k_inproj<<<(N_NODES * T_STEPS) / 8, 256, 0, stream>>>(
      x, sp_w, sp_b, sp_g, sp_be, ip_w, ip_b, ip_g, ip_be, e);
  k_conv_pool<<<N_NODES, 256, 0, stream>>>(
      e, c1_w, c1_b, bn1_g, bn1_b, bn1_m, bn1_v,
      c2_w, c2_b, bn2_g, bn2_b, bn2_m, bn2_v, dm, x_pool, xs);
  k_gcn_agg_wmma<<<32, 256, 0, stream>>>(Wm, xs, agg);
  k_gcn_post<<<N_NODES / 8, 256, 0, stream>>>(
      agg, x_pool, g1_w, g1_b, g1_g, g1_be, dm, h1, hs1);
  k_gcn_agg_wmma<<<32, 256, 0, stream>>>(Wm, hs1, agg);
  k_gcn_post<<<N_NODES / 8, 256, 0, stream>>>(
      agg, h1, g2_w, g2_b, g2_g, g2_be, dm, h2, nullptr);
  k_head<<<N_NODES / 8, 256, 0, stream>>>(
      x_pool, h2, t1_w, t1_b, t1_g, t1_be, t2_w, t2_b, t2_g, t2_be,
      mu_w, mu_b, lv_w, lv_b, (float*)d_out);
}